// ConvertDubaiMasks_2869038154238
// MI455X (gfx1250) — compile-verified
//
#include <hip/hip_runtime.h>
#include <stdint.h>

// ---------------------------------------------------------------------------
// Dubai RGB-mask -> class-id, MI455X (gfx1250).
// Memory-bound streaming kernel: TDM (tensor_load_to_lds) stages 3x256-pixel
// RGB tiles into wave-private LDS double buffers, pipelined on TENSORcnt;
// compute packs RGB into an exact f32 key (r + 256 g + 65536 b < 2^24) with
// two FMAs and does a 5-way v_cmp_eq_f32 / cndmask chain; results leave via
// non-temporal b128 stores (stream size >= 192 MB L2, so NT is correct).
// ---------------------------------------------------------------------------

typedef float    f32x4 __attribute__((ext_vector_type(4)));
typedef uint32_t u32x4 __attribute__((ext_vector_type(4)));
typedef uint32_t u32x8 __attribute__((ext_vector_type(8)));

#define TPB          256
#define NW           (TPB / 32)          // 8 wave32s per block
#define TILE_X       256                 // pixels per wave-stage (TDM tile width)
#define NSTAGE       2                   // double buffer per wave
#define LANE_F4      (TILE_X / 32 / 4)   // f32x4 chunks per lane per plane = 2
#define PX_PER_WAVE  (TILE_X * NSTAGE)   // 512
#define PX_PER_BLOCK (PX_PER_WAVE * NW)  // 4096
#define STAGE_F4     (3 * TILE_X / 4)    // f32x4 per stage buffer (3 planes) = 192

// Exact float key: r + 256*g + 65536*b (all integers < 2^24 -> exact in f32).
// Color 0 maps to id 0, same as "no match", so only colors 1..5 need compares.
#define FKEY(r, g, b) ((float)((r) + 256 * (g) + 65536 * (b)))

__device__ __forceinline__ float classify(float rf, float gf, float bf) {
  float key = fmaf(bf, 65536.0f, fmaf(gf, 256.0f, rf));  // 2 VALU, exact
  float id = 0.0f;
  id = (key == FKEY(132,  41, 246)) ? 1.0f : id;
  id = (key == FKEY(110, 193, 228)) ? 2.0f : id;
  id = (key == FKEY(254, 221,  58)) ? 3.0f : id;
  id = (key == FKEY(226, 169,  41)) ? 4.0f : id;
  id = (key == FKEY(155, 155, 155)) ? 5.0f : id;
  return id;
}

// One TDM descriptor: 2D tensor (dim0 = npix elements, dim1 = 3 planes,
// dim0_stride = npix), tile = TILE_X x 3, data_size = 4B, load to lds_addr.
__device__ __forceinline__ void tdm_load_tile(uint32_t lds_addr, uint64_t gaddr,
                                              uint32_t npix) {
  u32x4 g0;
  g0.x = 1u;                                                    // count=1 (valid user D#)
  g0.y = lds_addr;                                              // lds_addr[31:0]
  g0.z = (uint32_t)gaddr;                                       // global_addr[31:0]
  g0.w = ((uint32_t)(gaddr >> 32) & 0x01FFFFFFu) | (2u << 30);  // addr[56:32] | type=2
  u32x8 g1;
  g1[0] = 2u << 16;                        // workgroup_mask=0, data_size=2 (4B)
  g1[1] = (npix & 0xFFFFu) << 16;          // tensor_dim0[15:0]
  g1[2] = (npix >> 16) | (3u << 16);       // tensor_dim0[31:16] | tensor_dim1=3
  g1[3] = ((uint32_t)TILE_X) << 16;        // tile_dim0
  g1[4] = 3u;                              // tile_dim1=3, tile_dim2=0
  g1[5] = npix;                            // tensor_dim0_stride[31:0]
  g1[6] = 0u;                              // stride hi / dim1_stride lo
  g1[7] = 0u;                              // dim1_stride hi
  // 2D tensor -> descriptor groups 2/3 disabled (VADDR2/3 NULL form).
  asm volatile("tensor_load_to_lds %0, %1 th:TH_LOAD_NT"
               :: "s"(g0), "s"(g1) : "memory");
}

__global__ void __launch_bounds__(TPB)
dubai_masks_tdm(const float* __restrict__ t, float* __restrict__ out,
                uint32_t npix) {
  extern __shared__ f32x4 smem[];   // [NW][NSTAGE][3][TILE_X/4] = 48 KiB
  const uint32_t tid  = threadIdx.x;
  const uint32_t w    = tid >> 5;
  const uint32_t lane = tid & 31u;
  const uint32_t wave_base = (uint32_t)blockIdx.x * PX_PER_BLOCK + w * PX_PER_WAVE;

  f32x4* buf0 = smem + (w * NSTAGE + 0) * STAGE_F4;
  f32x4* buf1 = smem + (w * NSTAGE + 1) * STAGE_F4;

  // Issue both stage DMAs up front (wave-private buffers, no barriers needed;
  // TENSORcnt completes in order within a wave).
  tdm_load_tile((uint32_t)(uintptr_t)buf0,
                (uint64_t)(uintptr_t)(t + wave_base), npix);
  tdm_load_tile((uint32_t)(uintptr_t)buf1,
                (uint64_t)(uintptr_t)(t + wave_base + TILE_X), npix);

#pragma unroll
  for (int i = 0; i < NSTAGE; ++i) {
    if (i == 0) asm volatile("s_wait_tensorcnt 0x1" ::: "memory");
    else        asm volatile("s_wait_tensorcnt 0x0" ::: "memory");
    const f32x4* sb = (i == 0) ? buf0 : buf1;
    const f32x4* rp = sb;                       // plane 0 (R)
    const f32x4* gp = sb + (TILE_X / 4);        // plane 1 (G)
    const f32x4* bp = sb + 2 * (TILE_X / 4);    // plane 2 (B)
    const uint32_t pix0 = wave_base + (uint32_t)i * TILE_X + lane * (TILE_X / 32);
#pragma unroll
    for (int k = 0; k < LANE_F4; ++k) {
      f32x4 r = rp[lane * LANE_F4 + k];         // ds_load_b128
      f32x4 g = gp[lane * LANE_F4 + k];
      f32x4 b = bp[lane * LANE_F4 + k];
      f32x4 o;
      o.x = classify(r.x, g.x, b.x);
      o.y = classify(r.y, g.y, b.y);
      o.z = classify(r.z, g.z, b.z);
      o.w = classify(r.w, g.w, b.w);
      __builtin_nontemporal_store(o, (f32x4*)(out + pix0 + 4u * k));
    }
  }
}

// Scalar tail for pixel counts not divisible by PX_PER_BLOCK (unused at 4096^2).
__global__ void dubai_masks_tail(const float* __restrict__ t, float* __restrict__ out,
                                 uint32_t start, uint32_t npix) {
  uint32_t i = start + blockIdx.x * blockDim.x + threadIdx.x;
  if (i < npix) {
    out[i] = classify(t[i], t[i + (size_t)npix], t[i + 2 * (size_t)npix]);
  }
}

extern "C" void kernel_launch(void* const* d_in, const int* in_sizes, int n_in,
                              void* d_out, int out_size, void* d_ws, size_t ws_size,
                              hipStream_t stream) {
  (void)n_in; (void)out_size; (void)d_ws; (void)ws_size;
  const float* t = (const float*)d_in[0];   // (3, H, W) float32, plane-major
  float* out = (float*)d_out;               // (H, W) ids as float
  uint32_t npix = (uint32_t)(in_sizes[0] / 3);

  uint32_t nblocks = npix / PX_PER_BLOCK;
  size_t shmem = (size_t)NW * NSTAGE * STAGE_F4 * sizeof(f32x4);  // 49152 B
  if (nblocks) {
    dubai_masks_tdm<<<nblocks, TPB, shmem, stream>>>(t, out, npix);
  }
  uint32_t done = nblocks * PX_PER_BLOCK;
  if (done < npix) {
    dubai_masks_tail<<<(npix - done + 255) / 256, 256, 0, stream>>>(t, out, done, npix);
  }
}